// DropSAGE_49357764166329
// MI455X (gfx1250) — compile-verified
//
#include <hip/hip_runtime.h>
#include <hip/hip_bf16.h>
#include <math.h>

typedef __attribute__((ext_vector_type(2))) float v2f;
typedef __attribute__((ext_vector_type(8))) float v8f;

#define FH 64     // F == H == 64
#define ENCN 8
#define OUTN 2
#define BPITCH 80 // float2 pitch of B tiles in LDS: bank shift 32/row

// ---------------------------------------------------------------------------
// JAX threefry2x32 (key = PRNGKey(42) -> (0, 42)), iota counter split in
// halves as jax does for random_bits, then mantissa-bit uniform in [0,1).
// ---------------------------------------------------------------------------
__device__ __forceinline__ unsigned rotl32(unsigned x, int r) {
  return (x << r) | (x >> (32 - r));
}

__device__ __forceinline__ float jax_uniform_key42(unsigned idx, unsigned total) {
  const unsigned half = total >> 1;           // total (E) is even
  unsigned c0, c1;
  bool second;
  if (idx < half) { c0 = idx;        c1 = half + idx; second = false; }
  else            { c0 = idx - half; c1 = idx;        second = true;  }

  const unsigned ks0 = 0u, ks1 = 42u;
  const unsigned ks2 = 0x1BD11BDAu ^ ks0 ^ ks1;
  unsigned x0 = c0 + ks0, x1 = c1 + ks1;

  const int ra[4] = {13, 15, 26, 6};
  const int rb[4] = {17, 29, 16, 24};
#pragma unroll
  for (int i = 0; i < 4; i++) { x0 += x1; x1 = rotl32(x1, ra[i]); x1 ^= x0; }
  x0 += ks1; x1 += ks2 + 1u;
#pragma unroll
  for (int i = 0; i < 4; i++) { x0 += x1; x1 = rotl32(x1, rb[i]); x1 ^= x0; }
  x0 += ks2; x1 += ks0 + 2u;
#pragma unroll
  for (int i = 0; i < 4; i++) { x0 += x1; x1 = rotl32(x1, ra[i]); x1 ^= x0; }
  x0 += ks0; x1 += ks1 + 3u;
#pragma unroll
  for (int i = 0; i < 4; i++) { x0 += x1; x1 = rotl32(x1, rb[i]); x1 ^= x0; }
  x0 += ks1; x1 += ks2 + 4u;
#pragma unroll
  for (int i = 0; i < 4; i++) { x0 += x1; x1 = rotl32(x1, ra[i]); x1 ^= x0; }
  x0 += ks2; x1 += ks0 + 5u;

  unsigned bits = second ? x1 : x0;
  unsigned fb = (bits >> 9) | 0x3F800000u;
  return __uint_as_float(fb) - 1.0f;
}

// ---------------------------------------------------------------------------
// Degree counts (fp32: doubles as the segment-mean denominator; layer 2 uses
// the same counts because the reference counts all edges, dropped or not).
// ---------------------------------------------------------------------------
__global__ void count_kernel(const int* __restrict__ dst, float* __restrict__ cnt, int E) {
  int e = blockIdx.x * blockDim.x + threadIdx.x;
  if (e < E) atomicAdd(&cnt[dst[e]], 1.0f);
}

// ---------------------------------------------------------------------------
// Layer-1 scatter: agg1[dst] += x[src]. 16 lanes per edge, float4 per lane.
// ---------------------------------------------------------------------------
__global__ void scatter1_kernel(const float* __restrict__ x,
                                const int* __restrict__ src,
                                const int* __restrict__ dst,
                                float* __restrict__ agg, int E) {
  long long t = (long long)blockIdx.x * blockDim.x + threadIdx.x;
  int e = (int)(t >> 4);
  if (e >= E) return;
  int f4 = ((int)t & 15) * 4;
  int s = src[e], d = dst[e];
  const float4 v = *(const float4*)&x[(size_t)s * FH + f4];
  float* p = &agg[(size_t)d * FH + f4];
  atomicAdd(p + 0, v.x);
  atomicAdd(p + 1, v.y);
  atomicAdd(p + 2, v.z);
  atomicAdd(p + 3, v.w);
}

// ---------------------------------------------------------------------------
// Per-edge dropout decision (threefry once per edge, not 16x):
//   scale[e] = keep ? 1/(1-drop) : 0
// ---------------------------------------------------------------------------
__global__ void edge_scale_kernel(const float* __restrict__ lab,
                                  const int* __restrict__ src,
                                  const int* __restrict__ dst,
                                  float* __restrict__ scale, int E) {
  int e = blockIdx.x * blockDim.x + threadIdx.x;
  if (e >= E) return;
  float drop = fabsf(lab[dst[e]] - lab[src[e]]);
  float u = jax_uniform_key42((unsigned)e, (unsigned)E);
  scale[e] = (u >= drop) ? 1.0f / (1.0f - drop) : 0.0f;
}

// ---------------------------------------------------------------------------
// Layer-2 scatter with precomputed inverted-dropout scale.
// ---------------------------------------------------------------------------
__global__ void scatter2_kernel(const float* __restrict__ h,
                                const float* __restrict__ scale,
                                const int* __restrict__ src,
                                const int* __restrict__ dst,
                                float* __restrict__ agg, int E) {
  long long t = (long long)blockIdx.x * blockDim.x + threadIdx.x;
  int e = (int)(t >> 4);
  if (e >= E) return;
  float sc = scale[e];
  if (sc == 0.0f) return;                   // dropped edge contributes nothing
  int f4 = ((int)t & 15) * 4;
  int s = src[e], d = dst[e];
  const float4 v = *(const float4*)&h[(size_t)s * FH + f4];
  float* p = &agg[(size_t)d * FH + f4];
  atomicAdd(p + 0, v.x * sc);
  atomicAdd(p + 1, v.y * sc);
  atomicAdd(p + 2, v.z * sc);
  atomicAdd(p + 3, v.w * sc);
}

// ---------------------------------------------------------------------------
// SAGE dense stage, fp32 WMMA 16x16x4:
//   out = act( (agg / max(cnt,1)) @ Wl^T + bl + Aself @ Wr^T )
// One wave per 16-row tile, full 64 output columns (4 N-tiles).
// Fragment layouts per CDNA5 ISA 7.12.2 (f32):
//   A: lane holds row M = lane&15; VGPR v holds K = k0 + v + 2*(lane>>4)
//   B: lane holds col N = (lane&15)+16*nt; VGPR v holds K = k0 + v + 2*(lane>>4)
//   C/D: VGPR v holds row M = v + 8*(lane>>4), col N = lane&15 (+16*nt)
// B is stored k-pair interleaved in LDS (float2 per (kp,n)) so each fragment
// is a single naturally-aligned ds_load_b64 straight into the operand pair.
// ---------------------------------------------------------------------------
__global__ __launch_bounds__(256) void sage_gemm_kernel(
    const float* __restrict__ Aagg, const float* __restrict__ cnt,
    const float* __restrict__ Aself,
    const float* __restrict__ Wl, const float* __restrict__ bl,
    const float* __restrict__ Wr,
    float* __restrict__ out, int nRows, int nTiles, int do_relu) {
  // sB[mat][kp][n] = { W[n][2kp], W[n][2kp+1] }  (B = W^T)
  __shared__ float2 sB[2][FH / 2][BPITCH];

  for (int i = threadIdx.x; i < FH * FH; i += 256) {
    int n = i >> 6, k = i & 63;
    ((float*)&sB[0][k >> 1][n])[k & 1] = Wl[i];
    ((float*)&sB[1][k >> 1][n])[k & 1] = Wr[i];
  }
  __syncthreads();

  int wave = threadIdx.x >> 5;
  int lane = threadIdx.x & 31;
  int tile = blockIdx.x * 8 + wave;
  if (tile >= nTiles) return;

  int rowbase = tile * 16;
  int half = lane >> 4;            // 0 or 1
  int mrow = lane & 15;            // A row within tile / B,C column within ntile
  int row = rowbase + mrow;
  int rowc = row < nRows ? row : nRows - 1;   // clamp for ragged last tile

  float invc = 1.0f / fmaxf(cnt[rowc], 1.0f);

  v8f acc[4];
#pragma unroll
  for (int nt = 0; nt < 4; nt++) {
    float b = bl[nt * 16 + mrow];
    acc[nt] = (v8f){b, b, b, b, b, b, b, b};
  }

  const float* arowg = &Aagg[(size_t)rowc * FH];
  const float* arows = &Aself[(size_t)rowc * FH];

  for (int k0 = 0; k0 < FH; k0 += 4) {
    int ka = k0 + 2 * half;        // this lane's K pair base (even)
    int kp = ka >> 1;              // k-pair index into sB
    v2f ag = *(const v2f*)&arowg[ka];
    v2f as = *(const v2f*)&arows[ka];
    ag[0] *= invc;
    ag[1] *= invc;
#pragma unroll
    for (int nt = 0; nt < 4; nt++) {
      int n = nt * 16 + mrow;
      v2f bL = *(const v2f*)&sB[0][kp][n];
      v2f bR = *(const v2f*)&sB[1][kp][n];
      acc[nt] = __builtin_amdgcn_wmma_f32_16x16x4_f32(
          false, ag, false, bL, (short)0, acc[nt], false, false);
      acc[nt] = __builtin_amdgcn_wmma_f32_16x16x4_f32(
          false, as, false, bR, (short)0, acc[nt], false, false);
    }
  }

#pragma unroll
  for (int nt = 0; nt < 4; nt++) {
#pragma unroll
    for (int v = 0; v < 8; v++) {
      int r = rowbase + v + 8 * half;
      if (r < nRows) {
        float val = acc[nt][v];
        if (do_relu) val = fmaxf(val, 0.0f);
        out[(size_t)r * FH + nt * 16 + mrow] = val;
      }
    }
  }
}

// ---------------------------------------------------------------------------
// Per-node label score: lab = sigmoid(W_lsp . (W_pre @ h + b_pre) + b_lsp)
// ---------------------------------------------------------------------------
__global__ void label_kernel(const float* __restrict__ h,
                             const float* __restrict__ W_pre,
                             const float* __restrict__ b_pre,
                             const float* __restrict__ W_lsp,
                             const float* __restrict__ b_lsp,
                             float* __restrict__ lab, int N) {
  int n = blockIdx.x * blockDim.x + threadIdx.x;
  if (n >= N) return;
  const float* hr = &h[(size_t)n * FH];
  float acc = b_lsp[0];
#pragma unroll
  for (int e = 0; e < ENCN; e++) {
    float s = b_pre[e];
    for (int f = 0; f < FH; f++) s += W_pre[e * FH + f] * hr[f];
    acc += W_lsp[e] * s;
  }
  lab[n] = 1.0f / (1.0f + expf(-acc));
}

// ---------------------------------------------------------------------------
// Encoder + attention fusion + comb + log_softmax head. One thread per node;
// h1 tile + big weights staged in LDS for coalesced access.
// ---------------------------------------------------------------------------
__global__ __launch_bounds__(64) void fuse_kernel(
    const float* __restrict__ h1in,
    const float* __restrict__ tfeat, const float* __restrict__ dfeat,
    const float* __restrict__ W_tf, const float* __restrict__ b_tf,
    const float* __restrict__ W_deg, const float* __restrict__ b_deg,
    const float* __restrict__ W_enc, const float* __restrict__ b_enc,
    const float* __restrict__ W_x, const float* __restrict__ b_x,
    const float* __restrict__ W_comb, const float* __restrict__ b_comb,
    const float* __restrict__ W_out, const float* __restrict__ b_out,
    float* __restrict__ out_h1, float* __restrict__ out_lsm, int N) {
  __shared__ float sh[64][FH + 1];
  __shared__ float sWx[ENCN][FH];
  __shared__ float sWcomb[FH][FH + ENCN];
  __shared__ float sWenc[ENCN][ENCN];

  int base = blockIdx.x * 64;
  for (int i = threadIdx.x; i < 64 * FH; i += 64) {
    int nn = i >> 6, f = i & 63;
    int gn = base + nn;
    sh[nn][f] = (gn < N) ? h1in[(size_t)gn * FH + f] : 0.0f;
  }
  for (int i = threadIdx.x; i < ENCN * FH; i += 64) sWx[i >> 6][i & 63] = W_x[i];
  for (int i = threadIdx.x; i < FH * (FH + ENCN); i += 64)
    sWcomb[i / (FH + ENCN)][i % (FH + ENCN)] = W_comb[i];
  if (threadIdx.x < ENCN * ENCN)
    sWenc[threadIdx.x >> 3][threadIdx.x & 7] = W_enc[threadIdx.x];
  __syncthreads();

  int n = base + threadIdx.x;
  if (n >= N) return;

  float t = tfeat[n], dgr = dfeat[n];
  float tf[ENCN], dgv[ENCN];
#pragma unroll
  for (int e = 0; e < ENCN; e++) {
    tf[e] = W_tf[e] * t + b_tf[e];
    dgv[e] = W_deg[e] * dgr + b_deg[e];
  }

  float xp[ENCN];
#pragma unroll
  for (int e = 0; e < ENCN; e++) {
    float s = b_x[e];
    for (int f = 0; f < FH; f++) s += sWx[e][f] * sh[threadIdx.x][f];
    xp[e] = tanhf(s);
  }

  float sc0 = 0.0f, sc1 = 0.0f;
#pragma unroll
  for (int e = 0; e < ENCN; e++) {
    float p0 = b_enc[e], p1 = b_enc[e];
#pragma unroll
    for (int j = 0; j < ENCN; j++) {
      p0 += sWenc[e][j] * tf[j];
      p1 += sWenc[e][j] * dgv[j];
    }
    sc0 += tanhf(p0) * xp[e];
    sc1 += tanhf(p1) * xp[e];
  }
  float m = fmaxf(sc0, sc1);
  float e0 = expf(sc0 - m), e1 = expf(sc1 - m);
  float inv = 1.0f / (e0 + e1);
  float s0 = e0 * inv, s1 = e1 * inv;

  float ctx[ENCN];
#pragma unroll
  for (int e = 0; e < ENCN; e++) ctx[e] = tf[e] * s0 + dgv[e] * s1;

  float hn[FH];
  for (int o = 0; o < FH; o++) {
    float s = b_comb[o];
    for (int f = 0; f < FH; f++) s += sWcomb[o][f] * sh[threadIdx.x][f];
#pragma unroll
    for (int j = 0; j < ENCN; j++) s += sWcomb[o][FH + j] * ctx[j];
    hn[o] = s;
    out_h1[(size_t)n * FH + o] = s;
  }

  float l0 = b_out[0], l1 = b_out[1];
  for (int f = 0; f < FH; f++) {
    l0 += W_out[f] * hn[f];
    l1 += W_out[FH + f] * hn[f];
  }
  float mm = fmaxf(l0, l1);
  float lse = mm + logf(expf(l0 - mm) + expf(l1 - mm));
  out_lsm[(size_t)n * 2 + 0] = l0 - lse;
  out_lsm[(size_t)n * 2 + 1] = l1 - lse;
}

// ---------------------------------------------------------------------------
extern "C" void kernel_launch(void* const* d_in, const int* in_sizes, int n_in,
                              void* d_out, int out_size, void* d_ws, size_t ws_size,
                              hipStream_t stream) {
  const float* x     = (const float*)d_in[0];
  const int*   edge  = (const int*)d_in[1];
  const float* tfeat = (const float*)d_in[2];
  const float* dfeat = (const float*)d_in[3];
  const float* W1_l  = (const float*)d_in[4];
  const float* b1_l  = (const float*)d_in[5];
  const float* W1_r  = (const float*)d_in[6];
  const float* W2_l  = (const float*)d_in[7];
  const float* b2_l  = (const float*)d_in[8];
  const float* W2_r  = (const float*)d_in[9];
  const float* W_pre = (const float*)d_in[10];
  const float* b_pre = (const float*)d_in[11];
  const float* W_lsp = (const float*)d_in[12];
  const float* b_lsp = (const float*)d_in[13];
  const float* W_tf  = (const float*)d_in[14];
  const float* b_tf  = (const float*)d_in[15];
  const float* W_deg = (const float*)d_in[16];
  const float* b_deg = (const float*)d_in[17];
  const float* W_enc = (const float*)d_in[18];
  const float* b_enc = (const float*)d_in[19];
  const float* W_x   = (const float*)d_in[20];
  const float* b_x   = (const float*)d_in[21];
  const float* W_comb= (const float*)d_in[22];
  const float* b_comb= (const float*)d_in[23];
  const float* W_out = (const float*)d_in[24];
  const float* b_out = (const float*)d_in[25];

  const int N = in_sizes[0] / FH;
  const int E = in_sizes[1] / 2;
  const int* src = edge;
  const int* dst = edge + E;

  // workspace layout
  float* cnt  = (float*)d_ws;
  float* agg1 = cnt + N;
  float* h    = agg1 + (size_t)N * FH;
  float* lab  = h + (size_t)N * FH;
  float* agg2 = lab + N;
  float* h1   = agg2 + (size_t)N * FH;
  // agg1 is dead after the layer-1 GEMM; reuse its space for per-edge scales
  float* scale = agg1;             // E <= N*FH floats

  hipMemsetAsync(cnt,  0, (size_t)N * sizeof(float), stream);
  hipMemsetAsync(agg1, 0, (size_t)N * FH * sizeof(float), stream);
  hipMemsetAsync(agg2, 0, (size_t)N * FH * sizeof(float), stream);

  const int nTiles = (N + 15) / 16;
  const int gemmBlocks = (nTiles + 7) / 8;
  const long long scatterThreads = (long long)E * 16;
  const int scatterBlocks = (int)((scatterThreads + 255) / 256);

  count_kernel<<<(E + 255) / 256, 256, 0, stream>>>(dst, cnt, E);
  scatter1_kernel<<<scatterBlocks, 256, 0, stream>>>(x, src, dst, agg1, E);
  sage_gemm_kernel<<<gemmBlocks, 256, 0, stream>>>(agg1, cnt, x, W1_l, b1_l, W1_r,
                                                   h, N, nTiles, 1);
  label_kernel<<<(N + 255) / 256, 256, 0, stream>>>(h, W_pre, b_pre, W_lsp, b_lsp,
                                                    lab, N);
  edge_scale_kernel<<<(E + 255) / 256, 256, 0, stream>>>(lab, src, dst, scale, E);
  scatter2_kernel<<<scatterBlocks, 256, 0, stream>>>(h, scale, src, dst, agg2, E);
  sage_gemm_kernel<<<gemmBlocks, 256, 0, stream>>>(agg2, cnt, h, W2_l, b2_l, W2_r,
                                                   h1, N, nTiles, 0);

  float* out_h1 = (float*)d_out;
  float* out_lsm = out_h1 + (size_t)N * FH;
  fuse_kernel<<<(N + 63) / 64, 64, 0, stream>>>(h1, tfeat, dfeat,
                                                W_tf, b_tf, W_deg, b_deg,
                                                W_enc, b_enc, W_x, b_x,
                                                W_comb, b_comb, W_out, b_out,
                                                out_h1, out_lsm, N);
}